// BaseIODEModel_57664230916363
// MI455X (gfx1250) — compile-verified
//
#include <hip/hip_runtime.h>

// MI455X / gfx1250, wave32. fp32 WMMA (V_WMMA_F32_16X16X4_F32) path.
//
// dz = selfMLP(z) + sum_over_senders interMLP(pair features)
//   self:  81920 rows, 4->128->128->4
//   inter: 737280 rows, 6(pad 8)->128->128->4, 9-way sum per receiver
// ~29 GFLOP fp32 vs ~3MB traffic -> matrix-compute bound -> WMMA.
// M=32 rows (TILES=2) per wave iteration: each B-fragment ds_load_b64 feeds
// 2 WMMAs, keeping the WGP's shared LDS bandwidth below 4-SIMD compute demand.
// Layer-0 fragments + biases live in LDS; __launch_bounds__(128,1) lifts the
// default 1024-thread VGPR cap so the 128 accumulator VGPRs don't spill.

#define NBODY   10
#define F       4      // 2*NDIM
#define H       128
#define PAIRS   90     // NBODY*(NBODY-1)
#define NT      8      // H/16 N-tiles
#define KK      32     // H/4  K-steps
#define WAVES   4      // waves per block (blockDim = 128)
#define TILES   2      // 16-row tiles per wave iteration (M = 32)
#define HSTRIDE 132    // LDS row stride (floats), bank-conflict free

typedef __attribute__((ext_vector_type(2))) float v2f;
typedef __attribute__((ext_vector_type(4))) float v4f;
typedef __attribute__((ext_vector_type(8))) float v8f;

static __device__ __forceinline__ v8f wmma4(v2f a, v2f b, v8f c) {
  // D = A(16x4) * B(4x16) + C(16x16), fp32
  return __builtin_amdgcn_wmma_f32_16x16x4_f32(false, a, false, b, (short)0, c,
                                               false, false);
}

static __device__ __forceinline__ float softplus_f(float x) {
  // softplus(x) = max(x,0) + ln2 * log2(1 + exp2(-log2e*|x|))
  // 1+e is in [1,2]: raw v_log_f32 is full-accuracy there, no denorm guard needed.
  float e = __builtin_amdgcn_exp2f(-1.442695041f * fabsf(x));
  float l = __builtin_amdgcn_logf(1.0f + e);
  return fmaxf(x, 0.0f) + 0.6931471806f * l;
}

// Stage a 128x128 row-major weight into B-fragment order:
// dst[((kk*NT+t)*32 + lane)*2 + e] = W[kk*4 + 2*(lane>>4) + e][t*16 + (lane&15)]
static __device__ __forceinline__ void stage_w_full(float* dst,
                                                    const float* __restrict__ src,
                                                    int tid, int nthreads) {
  for (int i = tid; i < KK * NT * 64; i += nthreads) {
    int e    = i & 1;
    int lane = (i >> 1) & 31;
    int t    = (i >> 6) & 7;
    int kk   = i >> 9;
    int k    = kk * 4 + ((lane >> 4) << 1) + e;
    int n    = t * 16 + (lane & 15);
    dst[i]   = src[k * H + n];
  }
}

// Stage 128x4 output weight into a single padded 128x16 B-fragment tile.
static __device__ __forceinline__ void stage_w_out(float* dst,
                                                   const float* __restrict__ src,
                                                   int tid, int nthreads) {
  for (int i = tid; i < KK * 64; i += nthreads) {
    int e    = i & 1;
    int lane = (i >> 1) & 31;
    int kk   = i >> 6;
    int k    = kk * 4 + ((lane >> 4) << 1) + e;
    int n    = lane & 15;
    dst[i]   = (n < F) ? src[k * F + n] : 0.0f;
  }
}

// Stage layer-0 weight (kin x 128, kin <= 8) into 2 k-step B-fragment tiles,
// zero-padded beyond kin.
static __device__ __forceinline__ void stage_w_in(float* dst,
                                                  const float* __restrict__ src,
                                                  int kin, int tid, int nthreads) {
  for (int i = tid; i < 2 * NT * 64; i += nthreads) {
    int e    = i & 1;
    int lane = (i >> 1) & 31;
    int t    = (i >> 6) & 7;
    int kk   = i >> 9;
    int k    = kk * 4 + ((lane >> 4) << 1) + e;
    int n    = t * 16 + (lane & 15);
    dst[i]   = (k < kin) ? src[k * H + n] : 0.0f;
  }
}

// Layers 1 & 2 for TILES row-tiles. hb holds softplus(layer0) activations
// (TILES*16 x 128, stride HSTRIDE). c2[tt] cols >= F are garbage (caller masks).
static __device__ __forceinline__ void mlp_tail(const float* sW1f, const float* sW2f,
                                                const float* sB1, float* hb, int lane,
                                                v8f c2[TILES]) {
  const int h  = lane >> 4;
  const int nn = lane & 15;
  const v8f vzero = {0, 0, 0, 0, 0, 0, 0, 0};

  v8f c1[TILES][NT];
#pragma unroll
  for (int tt = 0; tt < TILES; ++tt)
#pragma unroll
    for (int t = 0; t < NT; ++t) c1[tt][t] = vzero;

  for (int kk = 0; kk < KK; ++kk) {
    v2f a[TILES];
#pragma unroll
    for (int tt = 0; tt < TILES; ++tt)
      a[tt] = *(const v2f*)&hb[(tt * 16 + nn) * HSTRIDE + kk * 4 + 2 * h];
    const float* bp = &sW1f[kk * (NT * 64) + lane * 2];
#pragma unroll
    for (int t = 0; t < NT; ++t) {
      v2f b = *(const v2f*)&bp[t * 64];   // one ds_load_b64 feeds TILES WMMAs
#pragma unroll
      for (int tt = 0; tt < TILES; ++tt) c1[tt][t] = wmma4(a[tt], b, c1[tt][t]);
    }
  }
  // bias + softplus, store H1 back (same-wave LDS ops are in-order; all H0
  // reads completed above, so in-place overwrite is safe)
#pragma unroll
  for (int t = 0; t < NT; ++t) {
    float b1v = sB1[t * 16 + nn];         // broadcast ds_load
#pragma unroll
    for (int tt = 0; tt < TILES; ++tt)
#pragma unroll
      for (int i = 0; i < 8; ++i)
        hb[(tt * 16 + h * 8 + i) * HSTRIDE + t * 16 + nn] =
            softplus_f(c1[tt][t][i] + b1v);
  }

#pragma unroll
  for (int tt = 0; tt < TILES; ++tt) c2[tt] = vzero;
  for (int kk = 0; kk < KK; ++kk) {
    v2f b = *(const v2f*)&sW2f[kk * 64 + lane * 2];
#pragma unroll
    for (int tt = 0; tt < TILES; ++tt) {
      v2f a = *(const v2f*)&hb[(tt * 16 + nn) * HSTRIDE + kk * 4 + 2 * h];
      c2[tt] = wmma4(a, b, c2[tt]);
    }
  }
}

__global__ void __launch_bounds__(WAVES * 32, 1)
self_mlp_kernel(const float* __restrict__ z,
                const float* __restrict__ fW0, const float* __restrict__ fb0,
                const float* __restrict__ fW1, const float* __restrict__ fb1,
                const float* __restrict__ fW2, const float* __restrict__ fb2,
                float* __restrict__ out, int nrows) {
  extern __shared__ float smem[];
  float* sW1f  = smem;                                  // 16384 floats
  float* sW2f  = sW1f + KK * NT * 64;                   //  2048 floats
  float* sW0f  = sW2f + KK * 64;                        //  1024 floats
  float* sB0   = sW0f + 2 * NT * 64;                    //   128
  float* sB1   = sB0 + H;                               //   128
  float* hbase = sB1 + H;                               // WAVES*TILES*16*HSTRIDE
  float* obase = hbase + WAVES * TILES * 16 * HSTRIDE;  // WAVES*TILES*64

  const int tid = threadIdx.x;
  stage_w_full(sW1f, fW1, tid, blockDim.x);
  stage_w_out(sW2f, fW2, tid, blockDim.x);
  stage_w_in(sW0f, fW0, F, tid, blockDim.x);            // K=4, kk=1 tile is zero
  for (int i = tid; i < H; i += blockDim.x) { sB0[i] = fb0[i]; sB1[i] = fb1[i]; }
  __syncthreads();

  const int lane = tid & 31;
  const int wv   = tid >> 5;
  const int h    = lane >> 4;
  const int nn   = lane & 15;
  float* hb = hbase + wv * TILES * 16 * HSTRIDE;
  float* ob = obase + wv * TILES * 64;

  const float b2r = (nn < F) ? fb2[nn] : 0.0f;
  const v8f vzero = {0, 0, 0, 0, 0, 0, 0, 0};

  const int ntiles = nrows / (16 * TILES);
  for (int tile = blockIdx.x * WAVES + wv; tile < ntiles; tile += gridDim.x * WAVES) {
    const int row0 = tile * 16 * TILES;
    // A0 frags: lane(m=nn, kpair=h) <- z[row][2h .. 2h+1]
    v2f a0[TILES];
#pragma unroll
    for (int tt = 0; tt < TILES; ++tt)
      a0[tt] = *(const v2f*)&z[(size_t)(row0 + tt * 16 + nn) * F + 2 * h];
    // layer 0 (K=4) -> softplus -> hb
#pragma unroll
    for (int t = 0; t < NT; ++t) {
      v2f   bw  = *(const v2f*)&sW0f[t * 64 + lane * 2];
      float b0v = sB0[t * 16 + nn];
#pragma unroll
      for (int tt = 0; tt < TILES; ++tt) {
        v8f c = wmma4(a0[tt], bw, vzero);
#pragma unroll
        for (int i = 0; i < 8; ++i)
          hb[(tt * 16 + h * 8 + i) * HSTRIDE + t * 16 + nn] =
              softplus_f(c[i] + b0v);
      }
    }
    v8f c2[TILES];
    mlp_tail(sW1f, sW2f, sB1, hb, lane, c2);
    // transpose 32x4 result through LDS -> coalesced b128 stores
    if (nn < F) {
#pragma unroll
      for (int tt = 0; tt < TILES; ++tt)
#pragma unroll
        for (int i = 0; i < 8; ++i)
          ob[(tt * 16 + h * 8 + i) * F + nn] = c2[tt][i] + b2r;
    }
    if (h == 0) {
#pragma unroll
      for (int tt = 0; tt < TILES; ++tt) {
        v4f v = *(const v4f*)&ob[(tt * 16 + nn) * F];
        *(v4f*)&out[(size_t)(row0 + tt * 16 + nn) * F] = v;
      }
    }
  }
}

__global__ void __launch_bounds__(WAVES * 32, 1)
inter_mlp_kernel(const float* __restrict__ z,
                 const float* __restrict__ iW0, const float* __restrict__ ib0,
                 const float* __restrict__ iW1, const float* __restrict__ ib1,
                 const float* __restrict__ iW2, const float* __restrict__ ib2,
                 float* __restrict__ out, int npairs) {
  extern __shared__ float smem[];
  float* sW1f  = smem;
  float* sW2f  = sW1f + KK * NT * 64;
  float* sW0f  = sW2f + KK * 64;
  float* sB0   = sW0f + 2 * NT * 64;
  float* sB1   = sB0 + H;
  float* hbase = sB1 + H;

  const int tid = threadIdx.x;
  stage_w_full(sW1f, iW1, tid, blockDim.x);
  stage_w_out(sW2f, iW2, tid, blockDim.x);
  stage_w_in(sW0f, iW0, 6, tid, blockDim.x);            // K=6 padded to 8
  for (int i = tid; i < H; i += blockDim.x) { sB0[i] = ib0[i]; sB1[i] = ib1[i]; }
  __syncthreads();

  const int lane = tid & 31;
  const int wv   = tid >> 5;
  const int h    = lane >> 4;
  const int nn   = lane & 15;
  float* hb = hbase + wv * TILES * 16 * HSTRIDE;

  const float b2r = (nn < F) ? ib2[nn] : 0.0f;
  const v8f vzero = {0, 0, 0, 0, 0, 0, 0, 0};

  const int ntiles = npairs / (16 * TILES);
  for (int tile = blockIdx.x * WAVES + wv; tile < ntiles; tile += gridDim.x * WAVES) {
    const int row0 = tile * 16 * TILES;
    // Build pair features; A-frags for k-steps 0,1: k = 4*kk + 2h + e
    v2f a0[TILES], a1[TILES];
#pragma unroll
    for (int tt = 0; tt < TILES; ++tt) {
      const int r   = row0 + tt * 16 + nn;
      const int g   = r / PAIRS;
      const int p   = r - g * PAIRS;
      const int rec = p / (NBODY - 1);
      const int q   = p - rec * (NBODY - 1);
      const int snd = q + (q >= rec ? 1 : 0);
      const float* zr = &z[(size_t)(g * NBODY + rec) * F];
      const float* zs = &z[(size_t)(g * NBODY + snd) * F];
      if (h == 0) {
        a0[tt] = v2f{zr[0] - zs[0], zr[1] - zs[1]};  // k=0,1
        a1[tt] = v2f{zs[2], zs[3]};                  // k=4,5
      } else {
        a0[tt] = v2f{zr[2], zr[3]};                  // k=2,3
        a1[tt] = v2f{0.0f, 0.0f};                    // k=6,7 (pad)
      }
    }
    // layer 0 (K=8) -> softplus -> hb
#pragma unroll
    for (int t = 0; t < NT; ++t) {
      v2f   bw0 = *(const v2f*)&sW0f[t * 64 + lane * 2];
      v2f   bw1 = *(const v2f*)&sW0f[(NT + t) * 64 + lane * 2];
      float b0v = sB0[t * 16 + nn];
#pragma unroll
      for (int tt = 0; tt < TILES; ++tt) {
        v8f c = wmma4(a0[tt], bw0, vzero);
        c     = wmma4(a1[tt], bw1, c);
#pragma unroll
        for (int i = 0; i < 8; ++i)
          hb[(tt * 16 + h * 8 + i) * HSTRIDE + t * 16 + nn] =
              softplus_f(c[i] + b0v);
      }
    }
    v8f c2[TILES];
    mlp_tail(sW1f, sW2f, sB1, hb, lane, c2);
    // 9-way receiver reduction via no-return global fp32 atomics
    if (nn < F) {
#pragma unroll
      for (int tt = 0; tt < TILES; ++tt)
#pragma unroll
        for (int i = 0; i < 8; ++i) {
          int r2   = row0 + tt * 16 + h * 8 + i;
          int g2   = r2 / PAIRS;
          int rec2 = (r2 - g2 * PAIRS) / (NBODY - 1);
          unsafeAtomicAdd(&out[(size_t)(g2 * NBODY + rec2) * F + nn], c2[tt][i] + b2r);
        }
    }
  }
}

extern "C" void kernel_launch(void* const* d_in, const int* in_sizes, int n_in,
                              void* d_out, int out_size, void* d_ws, size_t ws_size,
                              hipStream_t stream) {
  (void)n_in; (void)out_size; (void)d_ws; (void)ws_size;
  const float* z   = (const float*)d_in[0];
  const float* fW0 = (const float*)d_in[1];
  const float* fb0 = (const float*)d_in[2];
  const float* fW1 = (const float*)d_in[3];
  const float* fb1 = (const float*)d_in[4];
  const float* fW2 = (const float*)d_in[5];
  const float* fb2 = (const float*)d_in[6];
  const float* iW0 = (const float*)d_in[7];
  const float* ib0 = (const float*)d_in[8];
  const float* iW1 = (const float*)d_in[9];
  const float* ib1 = (const float*)d_in[10];
  const float* iW2 = (const float*)d_in[11];
  const float* ib2 = (const float*)d_in[12];
  float* out = (float*)d_out;

  const int nrows   = in_sizes[0] / F;        // 81920
  const int ngroups = nrows / NBODY;          // 8192
  const int npairs  = ngroups * PAIRS;        // 737280

  const size_t shmem =
      (size_t)(KK * NT * 64 + KK * 64 + 2 * NT * 64 + 2 * H +
               WAVES * TILES * 16 * HSTRIDE + WAVES * TILES * 64) * sizeof(float);

  const int t32_self = nrows / (16 * TILES);   // 2560
  const int t32_int  = npairs / (16 * TILES);  // 23040
  int grid_self = (t32_self + WAVES * 2 - 1) / (WAVES * 2);  // ~2 iters/wave
  int grid_int  = (t32_int + WAVES * 2 - 1) / (WAVES * 2);
  if (grid_self < 1) grid_self = 1;
  if (grid_int < 1) grid_int = 1;

  self_mlp_kernel<<<grid_self, WAVES * 32, shmem, stream>>>(
      z, fW0, fb0, fW1, fb1, fW2, fb2, out, nrows);
  inter_mlp_kernel<<<grid_int, WAVES * 32, shmem, stream>>>(
      z, iW0, ib0, iW1, ib1, iW2, ib2, out, npairs);
}